// GraphAndMessages_59957743452216
// MI455X (gfx1250) — compile-verified
//
#include <hip/hip_runtime.h>
#include <hip/hip_bf16.h>

typedef __attribute__((ext_vector_type(2))) float v2f;
typedef __attribute__((ext_vector_type(8))) float v8f;

#define Bn    64
#define Nn    128
#define Dn    32
#define NM1   127
#define Pn    (Nn * NM1)     // 16256
#define EPSc  1e-5f
#define SLOPEc 0.01f

__device__ __forceinline__ float leakyf(float v) { return v >= 0.f ? v : SLOPEc * v; }

__device__ __forceinline__ v8f wmma4(v2f a, v2f b, v8f c) {
  // V_WMMA_F32_16X16X4_F32 : D = A(16x4) * B(4x16) + C(16x16)
  return __builtin_amdgcn_wmma_f32_16x16x4_f32(
      /*neg_a=*/false, a, /*neg_b=*/false, b,
      /*c_mod=*/(short)0, c, /*reuse_a=*/false, /*reuse_b=*/false);
}

// ---------------------------------------------------------------------------
// Kernel 1: fused edge MLP.  One block per source node i; 8 waves per block
// (2 waves/SIMD -> large VGPR budget, no spills); each wave owns target
// indices jj = wave, wave+8, ...  All weight fragments and the j-invariant
// x_i A-fragments are pre-swizzled into LDS once per block, so WMMA operands
// stream from ds_load_b64 instead of scratch.
// ---------------------------------------------------------------------------
__global__ __launch_bounds__(256) void edge_mlp_kernel(
    const float* __restrict__ x,                          // (B,N,D)
    const float* __restrict__ W1, const float* __restrict__ b1,
    const float* __restrict__ g1, const float* __restrict__ be1,
    const float* __restrict__ W2, const float* __restrict__ b2,
    const float* __restrict__ W3, const float* __restrict__ b3,
    float* __restrict__ edges_out,                        // (B,P)
    float* __restrict__ agg_out)                          // (B,N,D)
{
  const int i    = blockIdx.x;
  const int tid  = threadIdx.x;
  const int wave = tid >> 5;       // 0..7
  const int lane = tid & 31;
  const int lh   = lane >> 4;      // half-wave: 0 => lanes 0-15, 1 => 16-31
  const int l16  = lane & 15;

  __shared__ float aggLDS[Bn][Dn];          // 8 KB
  __shared__ float scratch[8][16 * 32];     // 2 KB per wave, 16 KB total
  // Pre-swizzled per-lane operand fragments (shared by all waves):
  __shared__ v2f w1f[2 * 16 * 32];          // 8 KB  (nt,s,lane)
  __shared__ v2f w2f[2 * 8 * 32];           // 4 KB
  __shared__ v2f w3f[2 * 8 * 32];           // 4 KB
  __shared__ v2f w3c0f[8 * 32];             // 2 KB
  __shared__ v2f xif[4 * 8 * 32];           // 8 KB  (mt,s,lane) A-frags of x_i
  float* sc = scratch[wave];

  for (int t = tid; t < Bn * Dn; t += 256) aggLDS[t >> 5][t & 31] = 0.f;

  if (wave == 0) {
    // ---- stage swizzled weight fragments (lane formulas identical for all
    // waves, so wave 0's lanes can produce every fragment) ----
    // B layout (4x16): .x = row k (+2 for upper half-wave), .y = row k+1.
    #pragma unroll
    for (int nt = 0; nt < 2; ++nt) {
      const int n = nt * 16 + l16;
      #pragma unroll
      for (int s = 0; s < 16; ++s) {
        const int k = 4 * s + 2 * lh;
        v2f t; t.x = W1[k * 32 + n]; t.y = W1[(k + 1) * 32 + n];
        w1f[(nt * 16 + s) * 32 + lane] = t;
      }
      #pragma unroll
      for (int s = 0; s < 8; ++s) {
        const int k = 4 * s + 2 * lh;
        v2f t2; t2.x = W2[k * 32 + n]; t2.y = W2[(k + 1) * 32 + n];
        w2f[(nt * 8 + s) * 32 + lane] = t2;
        v2f t3; t3.x = W3[k * 33 + 1 + n]; t3.y = W3[(k + 1) * 33 + 1 + n];
        w3f[(nt * 8 + s) * 32 + lane] = t3;
      }
    }
    #pragma unroll
    for (int s = 0; s < 8; ++s) {
      const int k = 4 * s + 2 * lh;
      v2f t; t.x = W3[k * 33]; t.y = W3[(k + 1) * 33];
      w3c0f[s * 32 + lane] = t;
    }
    // ---- j-invariant A fragments: x[b, i, 0..31]  (row M = lane&15,
    // K = 4s + 2*(lane>=16), .y = K+1) ----
    #pragma unroll
    for (int mt = 0; mt < 4; ++mt) {
      const int b = mt * 16 + l16;
      const float* xi = &x[((size_t)b * Nn + i) * Dn];
      #pragma unroll
      for (int s = 0; s < 8; ++s)
        xif[(mt * 8 + s) * 32 + lane] = *(const v2f*)(xi + 4 * s + 2 * lh);
    }
  }

  // per-lane bias values (registers)
  float b1v[2], b2v[2], b3v[2];
  #pragma unroll
  for (int nt = 0; nt < 2; ++nt) {
    b1v[nt] = b1[nt * 16 + l16];
    b2v[nt] = b2[nt * 16 + l16];
    b3v[nt] = b3[1 + nt * 16 + l16];
  }
  const float b3_0 = b3[0];

  __syncthreads();

  // Persistent weighted-message accumulators, C layout (b = Mtile, d = Ntile).
  v8f agg[4][2];
  #pragma unroll
  for (int mt = 0; mt < 4; ++mt)
    #pragma unroll
    for (int nt = 0; nt < 2; ++nt) agg[mt][nt] = (v8f){};

  for (int jj = wave; jj < NM1; jj += 8) {       // wave-uniform trip count
    const int j = jj + (jj >= i ? 1 : 0);        // target node (skip i)
    const int p = i * NM1 + jj;                  // edge index
    const float gg = g1[p], bb = be1[p];

    // ---------------- Layer 1: h1 = [x_i | x_j] @ W1 + b1 ----------------
    v8f h1[4][2];
    #pragma unroll
    for (int mt = 0; mt < 4; ++mt) {
      h1[mt][0] = (v8f){};
      h1[mt][1] = (v8f){};
      const int b = mt * 16 + l16;
      const float* xj = &x[((size_t)b * Nn + j) * Dn];
      // prefetch the 8 j-dependent A fragments (K = 32..63) as one clause
      v2f Axj[8];
      #pragma unroll
      for (int s = 0; s < 8; ++s)
        Axj[s] = *(const v2f*)(xj + 4 * s + 2 * lh);
      #pragma unroll
      for (int s = 0; s < 8; ++s) {              // K = 0..31 (x_i half, LDS)
        const v2f a  = xif[(mt * 8 + s) * 32 + lane];
        h1[mt][0] = wmma4(a, w1f[s * 32 + lane],        h1[mt][0]);
        h1[mt][1] = wmma4(a, w1f[(16 + s) * 32 + lane], h1[mt][1]);
      }
      #pragma unroll
      for (int s = 8; s < 16; ++s) {             // K = 32..63 (x_j half)
        const v2f a = Axj[s - 8];
        h1[mt][0] = wmma4(a, w1f[s * 32 + lane],        h1[mt][0]);
        h1[mt][1] = wmma4(a, w1f[(16 + s) * 32 + lane], h1[mt][1]);
      }
    }

    // ---- per-edge normalization over (batch=64, hidden=32) = 2048 vals ----
    float ssum = 0.f, ssq = 0.f;
    #pragma unroll
    for (int mt = 0; mt < 4; ++mt)
      #pragma unroll
      for (int nt = 0; nt < 2; ++nt)
        #pragma unroll
        for (int e = 0; e < 8; ++e) {
          float v = h1[mt][nt][e] + b1v[nt];
          h1[mt][nt][e] = v;
          ssum += v; ssq += v * v;
        }
    #pragma unroll
    for (int off = 16; off > 0; off >>= 1) {
      ssum += __shfl_xor(ssum, off);
      ssq  += __shfl_xor(ssq,  off);
    }
    const float mean = ssum * (1.f / 2048.f);
    const float var  = ssq  * (1.f / 2048.f) - mean * mean;
    const float inv  = rsqrtf(var + EPSc);
    const float scl  = inv * gg;
    const float sft  = bb - mean * scl;
    #pragma unroll
    for (int mt = 0; mt < 4; ++mt)
      #pragma unroll
      for (int nt = 0; nt < 2; ++nt)
        #pragma unroll
        for (int e = 0; e < 8; ++e)
          h1[mt][nt][e] = leakyf(h1[mt][nt][e] * scl + sft);

    // ---------------- Layers 2 & 3 per 16-row M-tile ----------------
    #pragma unroll 1
    for (int mt = 0; mt < 4; ++mt) {
      // C-layout -> LDS (row-major 16x32) for A-layout reload (per-wave
      // scratch; same-wave LDS ops are in-order, no barrier needed).
      #pragma unroll
      for (int nt = 0; nt < 2; ++nt)
        #pragma unroll
        for (int e = 0; e < 8; ++e)
          sc[(e + 8 * lh) * 32 + nt * 16 + l16] = h1[mt][nt][e];

      v8f h2[2]; h2[0] = (v8f){}; h2[1] = (v8f){};
      #pragma unroll
      for (int s = 0; s < 8; ++s) {
        const v2f a = *(const v2f*)&sc[l16 * 32 + 4 * s + 2 * lh];
        h2[0] = wmma4(a, w2f[s * 32 + lane],       h2[0]);
        h2[1] = wmma4(a, w2f[(8 + s) * 32 + lane], h2[1]);
      }
      #pragma unroll
      for (int nt = 0; nt < 2; ++nt)
        #pragma unroll
        for (int e = 0; e < 8; ++e)
          sc[(e + 8 * lh) * 32 + nt * 16 + l16] = leakyf(h2[nt][e] + b2v[nt]);

      v8f h3[2]; h3[0] = (v8f){}; h3[1] = (v8f){};
      float lp = 0.f;                            // edge-logit partial (W3 col 0)
      #pragma unroll
      for (int s = 0; s < 8; ++s) {
        const v2f a = *(const v2f*)&sc[l16 * 32 + 4 * s + 2 * lh];
        h3[0] = wmma4(a, w3f[s * 32 + lane],       h3[0]);
        h3[1] = wmma4(a, w3f[(8 + s) * 32 + lane], h3[1]);
        const v2f c0 = w3c0f[s * 32 + lane];
        lp += a.x * c0.x + a.y * c0.y;
      }
      lp += __shfl_xor(lp, 16);                  // lane L holds logit of row L&15
      const float sig = 1.f / (1.f + __expf(-(lp + b3_0)));
      if (lh == 0)
        edges_out[(size_t)(mt * 16 + l16) * Pn + p] = sig;

      // weighted messages -> accumulate (C layout: row = e + 8*lh, d = lane&15)
      #pragma unroll
      for (int nt = 0; nt < 2; ++nt)
        #pragma unroll
        for (int e = 0; e < 8; ++e) {
          const float ev = __shfl(sig, e + 8 * lh);
          agg[mt][nt][e] += ev * (h3[nt][e] + b3v[nt]);
        }
    }
  }

  // ---- cross-wave reduction of message accumulators ----
  #pragma unroll
  for (int mt = 0; mt < 4; ++mt)
    #pragma unroll
    for (int nt = 0; nt < 2; ++nt)
      #pragma unroll
      for (int e = 0; e < 8; ++e)
        atomicAdd(&aggLDS[mt * 16 + e + 8 * lh][nt * 16 + l16], agg[mt][nt][e]);
  __syncthreads();

  for (int t = tid; t < Bn * Dn; t += 256) {
    const int b = t >> 5, d = t & 31;
    agg_out[((size_t)b * Nn + i) * Dn + d] = aggLDS[b][d];
  }
}

// ---------------------------------------------------------------------------
// Final node MLP (tiny: ~17 MFLOP).  Deterministic 3-stage split reduction.
// ---------------------------------------------------------------------------
__device__ __forceinline__ void compute_f_row(
    const float* __restrict__ x, const float* __restrict__ agg,
    const float* __restrict__ Wf1, const float* __restrict__ bf1,
    int r, float f[16])
{
  #pragma unroll
  for (int h = 0; h < 16; ++h) f[h] = bf1[h];
  const float* xr = &x[(size_t)r * 32];
  const float* ar = &agg[(size_t)r * 32];
  #pragma unroll 4
  for (int k = 0; k < 32; ++k) {
    const float v = xr[k];
    #pragma unroll
    for (int h = 0; h < 16; ++h) f[h] += v * Wf1[k * 16 + h];
  }
  #pragma unroll 4
  for (int k = 0; k < 32; ++k) {
    const float v = ar[k];
    #pragma unroll
    for (int h = 0; h < 16; ++h) f[h] += v * Wf1[(32 + k) * 16 + h];
  }
}

__global__ __launch_bounds__(128) void final_partial_kernel(
    const float* __restrict__ x, const float* __restrict__ agg,
    const float* __restrict__ Wf1, const float* __restrict__ bf1,
    float* __restrict__ partial)                 // (64, 32): [sum16 | sumsq16]
{
  __shared__ float ssum[16], ssq[16];
  const int tid = threadIdx.x;
  if (tid < 16) { ssum[tid] = 0.f; ssq[tid] = 0.f; }
  __syncthreads();

  const int r = blockIdx.x * 128 + tid;          // 64*128 = 8192 rows
  float f[16];
  compute_f_row(x, agg, Wf1, bf1, r, f);
  #pragma unroll
  for (int h = 0; h < 16; ++h) {
    atomicAdd(&ssum[h], f[h]);
    atomicAdd(&ssq[h],  f[h] * f[h]);
  }
  __syncthreads();
  if (tid < 16) {
    partial[blockIdx.x * 32 + tid]      = ssum[tid];
    partial[blockIdx.x * 32 + 16 + tid] = ssq[tid];
  }
}

__global__ void final_stats_kernel(
    const float* __restrict__ partial,
    const float* __restrict__ gf, const float* __restrict__ bef,
    float* __restrict__ ss)                      // [scale16 | shift16]
{
  const int h = threadIdx.x;
  if (h < 16) {
    float S = 0.f, Q = 0.f;
    for (int bl = 0; bl < 64; ++bl) {
      S += partial[bl * 32 + h];
      Q += partial[bl * 32 + 16 + h];
    }
    const float m   = S * (1.f / 8192.f);
    const float v   = Q * (1.f / 8192.f) - m * m;
    const float inv = rsqrtf(v + EPSc);
    const float scl = gf[h] * inv;
    ss[h]      = scl;
    ss[16 + h] = bef[h] - m * scl;
  }
}

__global__ __launch_bounds__(128) void final_out_kernel(
    const float* __restrict__ x, const float* __restrict__ agg,
    const float* __restrict__ Wf1, const float* __restrict__ bf1,
    const float* __restrict__ ss,
    const float* __restrict__ Wf2, const float* __restrict__ bf2,
    float* __restrict__ out)                     // (B*N, 32)
{
  const int r = blockIdx.x * 128 + threadIdx.x;
  float f[16];
  compute_f_row(x, agg, Wf1, bf1, r, f);
  float o[32];
  #pragma unroll
  for (int d = 0; d < 32; ++d) o[d] = bf2[d];
  #pragma unroll
  for (int h = 0; h < 16; ++h) {
    const float v = leakyf(f[h] * ss[h] + ss[16 + h]);
    #pragma unroll
    for (int d = 0; d < 32; ++d) o[d] += v * Wf2[h * 32 + d];
  }
  #pragma unroll
  for (int d = 0; d < 32; ++d) out[(size_t)r * 32 + d] = o[d];
}

extern "C" void kernel_launch(void* const* d_in, const int* in_sizes, int n_in,
                              void* d_out, int out_size, void* d_ws, size_t ws_size,
                              hipStream_t stream) {
  (void)in_sizes; (void)n_in; (void)out_size; (void)ws_size;
  const float* x   = (const float*)d_in[0];
  // d_in[1]=src, d_in[2]=tgt: structure is known analytically, unused.
  const float* W1  = (const float*)d_in[3];
  const float* b1  = (const float*)d_in[4];
  const float* g1  = (const float*)d_in[5];
  const float* be1 = (const float*)d_in[6];
  const float* W2  = (const float*)d_in[7];
  const float* b2  = (const float*)d_in[8];
  const float* W3  = (const float*)d_in[9];
  const float* b3  = (const float*)d_in[10];
  const float* Wf1 = (const float*)d_in[11];
  const float* bf1 = (const float*)d_in[12];
  const float* gf  = (const float*)d_in[13];
  const float* bef = (const float*)d_in[14];
  const float* Wf2 = (const float*)d_in[15];
  const float* bf2 = (const float*)d_in[16];

  float* edges = (float*)d_out;                       // (B,P)
  float* outp  = (float*)d_out + (size_t)Bn * Pn;     // (B,N,D)

  float* agg     = (float*)d_ws;                      // B*N*D   = 262144 f
  float* partial = agg + (size_t)Bn * Nn * Dn;        // 64*32   = 2048 f
  float* ss      = partial + 64 * 32;                 // 32 f

  edge_mlp_kernel<<<Nn, 256, 0, stream>>>(x, W1, b1, g1, be1, W2, b2, W3, b3,
                                          edges, agg);
  final_partial_kernel<<<64, 128, 0, stream>>>(x, agg, Wf1, bf1, partial);
  final_stats_kernel<<<1, 16, 0, stream>>>(partial, gf, bef, ss);
  final_out_kernel<<<64, 128, 0, stream>>>(x, agg, Wf1, bf1, ss, Wf2, bf2, outp);
}